// EntityDiffAttnLayer_17231408792038
// MI455X (gfx1250) — compile-verified
//
#include <hip/hip_runtime.h>
#include <hip/hip_bf16.h>

typedef __attribute__((ext_vector_type(2))) float v2f;
typedef __attribute__((ext_vector_type(8))) float v8f;

#define B_TOT 4096
#define E_ 128
#define A_ 16
#define D_ 128
#define H_ 4
#define HD_ 16
#define LDE 132                       // padded LDS row stride (bank-conflict free)
#define LAMBDA_INIT_F 0.35550906765887f   // 0.8 - 0.6*exp(-0.3)
#define RMS_EPS_F 1e-5f

static __device__ __forceinline__ v8f wmma4(v2f a, v2f b, v8f c) {
  // V_WMMA_F32_16X16X4_F32 : D = A(16x4) * B(4x16) + C(16x16)
  return __builtin_amdgcn_wmma_f32_16x16x4_f32(false, a, false, b, (short)0, c,
                                               false, false);
}

__global__ __launch_bounds__(256, 1)
void diff_attn_kernel(const float* __restrict__ ent,
                      const int* __restrict__ obs_mask,
                      const int* __restrict__ agent_mask,
                      const float* __restrict__ Wq, const float* __restrict__ Wk,
                      const float* __restrict__ Wv, const float* __restrict__ Wo,
                      const float* __restrict__ lq1, const float* __restrict__ lk1,
                      const float* __restrict__ lq2, const float* __restrict__ lk2,
                      const float* __restrict__ rms_w,
                      float* __restrict__ out_attn, float* __restrict__ out_aw) {
  __shared__ float sEnt[E_ * LDE];   // entities; reused as scores S (8h x 16a rows)
  __shared__ float sK[E_ * LDE];     // K; reused as AW (4h x 16a rows)
  __shared__ float sV[E_ * LDE];     // V
  __shared__ float sQ[A_ * LDE];     // Q (pre-scaled)
  __shared__ float sX[A_ * LDE];     // attn pre-out-proj (a x 128)
  __shared__ int   sMask[A_ * E_];

  const int b    = blockIdx.x;
  const int tid  = threadIdx.x;
  const int lane = tid & 31;
  const int wave = tid >> 5;    // 0..7
  const int half = lane >> 4;   // 0/1 (K-pair select per WMMA fragment layout)
  const int l16  = lane & 15;

  // lambda scalar (tiny dot products; L2-resident)
  float s1 = 0.f, s2 = 0.f;
  for (int i = 0; i < HD_; ++i) { s1 += lq1[i] * lk1[i]; s2 += lq2[i] * lk2[i]; }
  const float lam = __expf(s1) - __expf(s2) + LAMBDA_INIT_F;

  // ---- Phase 1: entities + obs_mask -> LDS ----
  const float* entB = ent + (size_t)b * E_ * D_;
  for (int i = tid; i < E_ * D_ / 4; i += 256) {
    int row = i >> 5;
    int col = (i & 31) << 2;
    float4 v4 = *(const float4*)(entB + row * D_ + col);
    float* d = &sEnt[row * LDE + col];
    d[0] = v4.x; d[1] = v4.y; d[2] = v4.z; d[3] = v4.w;
  }
  for (int i = tid; i < A_ * E_; i += 256)
    sMask[i] = obs_mask[(size_t)b * A_ * E_ + i];
  __syncthreads();

  // ---- Phase 2: Q = ent[0:16]*Wq^T*0.25 ; K = ent*Wk^T ; V = ent*Wv^T ----
  // 64 + 64 + 8 = 136 output tiles of 16x16, K-dim 128 (32 wmma steps each)
  for (int t = wave; t < 136; t += 8) {
    const float* Wm; float* dst; int mb, nb; float scale;
    if (t < 64)       {            Wm = Wk; dst = sK; mb = (t >> 3) << 4; nb = (t & 7) << 4; scale = 1.f; }
    else if (t < 128) { int u = t - 64;  Wm = Wv; dst = sV; mb = (u >> 3) << 4; nb = (u & 7) << 4; scale = 1.f; }
    else              { int u = t - 128; Wm = Wq; dst = sQ; mb = 0;            nb = u << 4;        scale = 0.25f; }
    v8f acc = {0.f, 0.f, 0.f, 0.f, 0.f, 0.f, 0.f, 0.f};
    const float* arow = &sEnt[(mb + l16) * LDE + 2 * half];   // A: entity rows
    const float* brow = &Wm[(nb + l16) * (size_t)D_ + 2 * half]; // B: weight rows (=> W^T)
#pragma unroll
    for (int k = 0; k < D_; k += 4) {
      v2f av = *(const v2f*)(arow + k);
      v2f bv = *(const v2f*)(brow + k);
      acc = wmma4(av, bv, acc);
    }
#pragma unroll
    for (int r = 0; r < 8; ++r)
      dst[(mb + r + 8 * half) * LDE + nb + l16] = acc[r] * scale;
  }
  __syncthreads();

  // ---- Phase 3: scores S_h = Q_h (16x16) * K_h^T (16x128), 8 half-heads ----
  for (int t = wave; t < 64; t += 8) {
    int h  = t >> 3;
    int eb = (t & 7) << 4;
    v8f acc = {0.f, 0.f, 0.f, 0.f, 0.f, 0.f, 0.f, 0.f};
    const float* arow = &sQ[l16 * LDE + h * HD_ + 2 * half];
    const float* brow = &sK[(eb + l16) * LDE + h * HD_ + 2 * half];
#pragma unroll
    for (int k = 0; k < HD_; k += 4)
      acc = wmma4(*(const v2f*)(arow + k), *(const v2f*)(brow + k), acc);
#pragma unroll
    for (int r = 0; r < 8; ++r)
      sEnt[(h * A_ + r + 8 * half) * LDE + eb + l16] = acc[r];  // sEnt reused as S
  }
  __syncthreads();

  // ---- Phase 4a: masked softmax over E per (half-head, agent) row ----
  if (tid < 128) {
    int h = tid >> 4, a = tid & 15;
    float* row = &sEnt[(h * A_ + a) * LDE];
    const int* mrow = &sMask[a * E_];
    float mx = -__builtin_inff();
    for (int e = 0; e < E_; ++e) {
      float s = mrow[e] ? -__builtin_inff() : row[e];
      row[e] = s;
      mx = fmaxf(mx, s);
    }
    if (mx == -__builtin_inff()) {            // fully-masked row -> nan_to_num(0)
      for (int e = 0; e < E_; ++e) row[e] = 0.f;
    } else {
      float sum = 0.f;
      for (int e = 0; e < E_; ++e) { float w = __expf(row[e] - mx); row[e] = w; sum += w; }
      float inv = 1.f / sum;
      for (int e = 0; e < E_; ++e) row[e] *= inv;
    }
  }
  __syncthreads();

  // ---- Phase 4b: AW = w0 - lam*w1 -> LDS (sK reuse) + coalesced HBM store ----
  float* awg = out_aw + (size_t)b * (H_ * A_ * E_);
  for (int i = tid; i < H_ * A_ * E_; i += 256) {
    int e = i & (E_ - 1);
    int rowi = i >> 7;              // hh*16 + a
    int hh = rowi >> 4, a = rowi & 15;
    float v = sEnt[(hh * 32 + a) * LDE + e] - lam * sEnt[(hh * 32 + 16 + a) * LDE + e];
    sK[rowi * LDE + e] = v;
    awg[i] = v;
  }
  __syncthreads();

  // ---- Phase 5: attn = AW_h (16x128) * V_h (128x32); 8 tiles, one per wave ----
  {
    int hh = wave >> 1, nt = wave & 1;
    int cb = hh * 32 + nt * 16;
    v8f acc = {0.f, 0.f, 0.f, 0.f, 0.f, 0.f, 0.f, 0.f};
    const float* arow = &sK[(hh * A_ + l16) * LDE + 2 * half];
    const float* bcol = &sV[(2 * half) * LDE + cb + l16];   // B indexed [k'=e][n=col]
#pragma unroll
    for (int k = 0; k < E_; k += 4) {
      v2f av = *(const v2f*)(arow + k);
      v2f bv; bv.x = bcol[k * LDE]; bv.y = bcol[k * LDE + LDE];
      acc = wmma4(av, bv, acc);
    }
#pragma unroll
    for (int r = 0; r < 8; ++r)
      sX[(r + 8 * half) * LDE + cb + l16] = acc[r];
  }
  __syncthreads();

  // ---- Phase 6: RMS norm over 32-dim per (agent, head) + rms_w * (1-lambda_init) ----
  if (tid < 64) {
    int a = tid >> 2, hh = tid & 3;
    float* xr = &sX[a * LDE + hh * 32];
    float ss = 0.f;
    for (int j = 0; j < 32; ++j) ss += xr[j] * xr[j];
    float sc = rsqrtf(ss * (1.f / 32.f) + RMS_EPS_F) * (1.f - LAMBDA_INIT_F);
    for (int j = 0; j < 32; ++j) xr[j] = xr[j] * sc * rms_w[j];
  }
  __syncthreads();

  // ---- Phase 7: out = X (16x128) * Wo^T, agent mask, write HBM ----
  {
    int nb = wave << 4;
    v8f acc = {0.f, 0.f, 0.f, 0.f, 0.f, 0.f, 0.f, 0.f};
    const float* arow = &sX[l16 * LDE + 2 * half];
    const float* brow = &Wo[(nb + l16) * (size_t)D_ + 2 * half];
#pragma unroll
    for (int k = 0; k < D_; k += 4)
      acc = wmma4(*(const v2f*)(arow + k), *(const v2f*)(brow + k), acc);
    float* og = out_attn + (size_t)b * A_ * D_;
#pragma unroll
    for (int r = 0; r < 8; ++r) {
      int a = r + 8 * half;
      float m = agent_mask[b * A_ + a] ? 0.f : 1.f;
      og[a * D_ + nb + l16] = acc[r] * m;
    }
  }
}

extern "C" void kernel_launch(void* const* d_in, const int* in_sizes, int n_in,
                              void* d_out, int out_size, void* d_ws, size_t ws_size,
                              hipStream_t stream) {
  const float* ent        = (const float*)d_in[0];
  const int*   obs_mask   = (const int*)d_in[1];
  const int*   agent_mask = (const int*)d_in[2];
  const float* Wq         = (const float*)d_in[3];
  const float* Wk         = (const float*)d_in[4];
  const float* Wv         = (const float*)d_in[5];
  const float* Wo         = (const float*)d_in[6];
  const float* lq1        = (const float*)d_in[7];
  const float* lk1        = (const float*)d_in[8];
  const float* lq2        = (const float*)d_in[9];
  const float* lk2        = (const float*)d_in[10];
  const float* rms_w      = (const float*)d_in[11];
  float* out_attn = (float*)d_out;
  float* out_aw   = out_attn + (size_t)B_TOT * A_ * D_;   // outputs concatenated flat

  diff_attn_kernel<<<B_TOT, 256, 0, stream>>>(ent, obs_mask, agent_mask,
                                              Wq, Wk, Wv, Wo,
                                              lq1, lk1, lq2, lk2, rms_w,
                                              out_attn, out_aw);
}